// MedianPool2d_7078106104315
// MI455X (gfx1250) — compile-verified
//
#include <hip/hip_runtime.h>
#include <hip/hip_bf16.h>

// ---------------------------------------------------------------------------
// 3x3 median filter, reflect padding, fp32, shape [8,3,1024,1024].
// Memory-bound: ~201 MB total traffic -> ~8.6 us floor at 23.3 TB/s.
// Strategy: async global->LDS staging (CDNA5 ASYNCcnt path), each input
// element read from HBM exactly once; column min/med/max decomposition for
// ~21 VALU min/max ops per pixel.
// ---------------------------------------------------------------------------

#define TW       128          // tile width  (outputs)
#define TH       32           // tile height (outputs)
#define LDSW     132          // TW+2 padded to keep float4 slices 16B aligned
#define LDSH     34           // TH+2
#define NTHREADS 256

// ---- CDNA5 async global->LDS copy (guarded; falls back to plain copy) -----
#if defined(__has_builtin)
#  if __has_builtin(__builtin_amdgcn_global_load_async_to_lds_b32)
#    define HAVE_ASYNC_LDS 1
#  endif
#  if __has_builtin(__builtin_amdgcn_s_wait_asynccnt)
#    define HAVE_WAIT_ASYNC 1
#  endif
#endif

typedef __attribute__((address_space(1))) int   as1_int;
typedef __attribute__((address_space(3))) int   as3_int;
typedef __attribute__((address_space(1))) const float as1_cfloat;
typedef __attribute__((address_space(3))) float as3_float;

__device__ __forceinline__ void async_copy_f32(const float* gp, float* lp) {
#ifdef HAVE_ASYNC_LDS
  as1_cfloat* g1 = (as1_cfloat*)gp;   // generic -> global addrspace
  as3_float*  l3 = (as3_float*)lp;    // generic -> LDS addrspace
  __builtin_amdgcn_global_load_async_to_lds_b32(
      (as1_int*)g1,                   // reinterpret within AS1 (drops const)
      (as3_int*)l3,                   // reinterpret within AS3
      /*offset=*/0, /*cpol=*/0);
#else
  *lp = *gp;   // fallback: global_load + ds_store
#endif
}

__device__ __forceinline__ void async_wait_all() {
#ifdef HAVE_ASYNC_LDS
#  ifdef HAVE_WAIT_ASYNC
  __builtin_amdgcn_s_wait_asynccnt(0);
#  else
  asm volatile("s_wait_asynccnt 0" ::: "memory");
#  endif
#endif
}

// reflect index for jnp.pad(..., mode='reflect'): -1 -> 1, N -> N-2, etc.
__device__ __forceinline__ int reflect_idx(int i, int n) {
  i = (i < 0) ? -i : i;
  i = (i >= n) ? (2 * n - 2 - i) : i;
  return i;
}

__global__ __launch_bounds__(NTHREADS)
void median3x3_kernel(const float* __restrict__ x, float* __restrict__ out,
                      int H, int W) {
  __shared__ __align__(16) float tile[LDSH * LDSW];

  const int    tid   = threadIdx.x;
  const int    c0    = blockIdx.x * TW;
  const int    r0    = blockIdx.y * TH;
  const size_t plane = (size_t)blockIdx.z * (size_t)H * (size_t)W;
  const float* src   = x + plane;

  // ---- stage halo tile [r0-1 .. r0+TH] x [c0-1 .. c0+LDSW-2] into LDS ----
  const int TOT = LDSH * LDSW;
  const bool interior = (r0 >= 1) && (r0 + LDSH - 1 <= H) &&
                        (c0 >= 1) && (c0 + LDSW - 1 <= W);
  if (interior) {
    const float* base = src + (size_t)(r0 - 1) * W + (c0 - 1);
    for (int i = tid; i < TOT; i += NTHREADS) {
      const int row = i / LDSW;
      const int col = i - row * LDSW;
      async_copy_f32(base + (size_t)row * W + col, &tile[i]);
    }
  } else {
    for (int i = tid; i < TOT; i += NTHREADS) {
      const int row = i / LDSW;
      const int col = i - row * LDSW;
      const int gr  = reflect_idx(r0 - 1 + row, H);
      const int gc  = reflect_idx(c0 - 1 + col, W);
      async_copy_f32(src + (size_t)gr * W + gc, &tile[i]);
    }
  }
  async_wait_all();
  __syncthreads();

  // ---- compute: each thread -> 4-wide run, 4 rows (stride 8) -------------
  const int txi = tid & 31;     // 32 runs of 4 px cover 128 columns
  const int tyi = tid >> 5;     // 8 row-groups
  const int xb  = txi << 2;     // tile-local output column base (16B aligned)
  float* dst = out + plane;

#pragma unroll
  for (int yy = 0; yy < TH / 8; ++yy) {
    const int oy = tyi + yy * 8;                 // tile-local output row
    const float* p0 = &tile[oy * LDSW + xb];     // LDS rows oy, oy+1, oy+2
    const float* p1 = p0 + LDSW;
    const float* p2 = p1 + LDSW;

    const float4 a4 = *(const float4*)p0; const float2 a2 = *(const float2*)(p0 + 4);
    const float4 b4 = *(const float4*)p1; const float2 b2 = *(const float2*)(p1 + 4);
    const float4 c4 = *(const float4*)p2; const float2 c2 = *(const float2*)(p2 + 4);

    const float A[6]  = {a4.x, a4.y, a4.z, a4.w, a2.x, a2.y};
    const float Bv[6] = {b4.x, b4.y, b4.z, b4.w, b2.x, b2.y};
    const float Cv[6] = {c4.x, c4.y, c4.z, c4.w, c2.x, c2.y};

    // vertical sort of each of the 6 columns: lo/mid/hi (6 ops each)
    float lo[6], mi[6], hi[6];
#pragma unroll
    for (int j = 0; j < 6; ++j) {
      const float l01 = fminf(A[j], Bv[j]);
      const float h01 = fmaxf(A[j], Bv[j]);
      lo[j] = fminf(l01, Cv[j]);
      hi[j] = fmaxf(h01, Cv[j]);
      mi[j] = fmaxf(l01, fminf(h01, Cv[j]));
    }

    // med9 = med3( max3(column mins), med3(column meds), min3(column maxes) )
    float r[4];
#pragma unroll
    for (int i = 0; i < 4; ++i) {
      const float mx  = fmaxf(fmaxf(lo[i], lo[i + 1]), lo[i + 2]);
      const float mn  = fminf(fminf(hi[i], hi[i + 1]), hi[i + 2]);
      const float l01 = fminf(mi[i], mi[i + 1]);
      const float h01 = fmaxf(mi[i], mi[i + 1]);
      const float md  = fmaxf(l01, fminf(h01, mi[i + 2]));
      const float l2  = fminf(mx, md);
      const float h2  = fmaxf(mx, md);
      r[i] = fmaxf(l2, fminf(h2, mn));
    }

    *(float4*)(dst + (size_t)(r0 + oy) * W + (c0 + xb)) =
        make_float4(r[0], r[1], r[2], r[3]);
  }
}

extern "C" void kernel_launch(void* const* d_in, const int* in_sizes, int n_in,
                              void* d_out, int out_size, void* d_ws, size_t ws_size,
                              hipStream_t stream) {
  (void)in_sizes; (void)n_in; (void)d_ws; (void)ws_size; (void)out_size;
  const float* x   = (const float*)d_in[0];
  float*       out = (float*)d_out;

  const int B = 8, C = 3, H = 1024, W = 1024;
  dim3 grid(W / TW, H / TH, B * C);   // 8 x 32 x 24 = 6144 blocks
  dim3 block(NTHREADS, 1, 1);
  median3x3_kernel<<<grid, block, 0, stream>>>(x, out, H, W);
}